// select_71708773974735
// MI455X (gfx1250) — compile-verified
//
#include <hip/hip_runtime.h>

// ---------------------------------------------------------------------------
// Shapes (hardcoded from the reference): bz=4, C=32, H=W=256, NF=12
// Phase 1: 48 slice-MAE partial sums (memory-bound stream, ~417 MB @ 23.3 TB/s)
// Phase 2: deterministic reduce + 12x12 pairwise-argmax per batch (tiny)
// Phase 3: gather two selected slices per batch -> [2*bz, C, H, W] (128 MB)
// ---------------------------------------------------------------------------

typedef float f4  __attribute__((ext_vector_type(4)));
typedef float v2f __attribute__((ext_vector_type(2)));
typedef float v8f __attribute__((ext_vector_type(8)));

#define NF      12
#define BZ      4
#define NCH     32
#define HW      65536               // 256*256
#define CHW     (NCH * HW)          // 2097152 floats per slice
#define CHW4    (CHW / 4)
#define THREADS 256
#define ITER    (HW / (THREADS * 4))  // 64: one channel plane per block
#define NSLICE  (BZ * NF)           // 48
#define NCHUNK  NCH                 // 32 chunks (channels) per slice

// ---------------------------------------------------------------------------
// Stage 1: each block reduces one channel plane (64K floats) of one (b,f)
// slice: sum |sal * (x - x_focal)|.  x_focal is streamed non-temporally (384MB
// once); x (32MB) and sal (1MB) stay resident in the 192MB L2 across the 12
// focal reuses.  Wave reduction uses V_WMMA_F32_16X16X4_F32 with B = ones:
// A layout (16x4 f32): lane L<16 holds A[L, 0..1], lane L>=16 holds A[L-16, 2..3]
// -> with a = {p, 0}:  D[m,n] = p_m + p_{m+16}  for every column n.
// D layout: lane n<16 holds rows 0..7, lane n>=16 holds rows 8..15, so
// sum_r c[r] (low lane) + sum_r c[r] (high lane) = total wave sum.
// ---------------------------------------------------------------------------
__global__ __launch_bounds__(THREADS) void mae_partial_kernel(
    const float* __restrict__ x, const float* __restrict__ xf,
    const float* __restrict__ sal, float* __restrict__ partials)
{
    const int bid = blockIdx.x;
    const int s   = bid / NCHUNK;   // slice index = b*NF + f
    const int c   = bid % NCHUNK;   // channel (chunk)
    const int b   = s / NF;
    const int f   = s % NF;

    const f4* __restrict__ px  = (const f4*)(x   + (size_t)b * CHW + (size_t)c * HW);
    const f4* __restrict__ pxf = (const f4*)(xf  + (size_t)(f * BZ + b) * CHW + (size_t)c * HW);
    const f4* __restrict__ ps  = (const f4*)(sal + (size_t)b * HW);

    const int t = threadIdx.x;
    float acc = 0.0f;

#pragma unroll 4
    for (int it = 0; it < ITER; ++it) {
        const int i = it * THREADS + t;              // coalesced float4 index
        f4 vx = px[i];
        f4 vf = __builtin_nontemporal_load(pxf + i); // streaming: keep L2 for x/sal
        f4 vs = ps[i];
        acc += __builtin_fabsf(vs[0] * (vx[0] - vf[0]));
        acc += __builtin_fabsf(vs[1] * (vx[1] - vf[1]));
        acc += __builtin_fabsf(vs[2] * (vx[2] - vf[2]));
        acc += __builtin_fabsf(vs[3] * (vx[3] - vf[3]));
    }

    // ---- wave32 reduction via WMMA (EXEC is all-1s here: no divergence) ----
    v2f a;     a[0] = acc;  a[1] = 0.0f;
    v2f bones; bones[0] = 1.0f; bones[1] = 1.0f;
    v8f cacc = {};
    cacc = __builtin_amdgcn_wmma_f32_16x16x4_f32(
        /*neg_a=*/false, a, /*neg_b=*/false, bones,
        /*c_mod=*/(short)0, cacc, /*reuse_a=*/false, /*reuse_b=*/false);
    float wsumv = cacc[0] + cacc[1] + cacc[2] + cacc[3]
                + cacc[4] + cacc[5] + cacc[6] + cacc[7];
    wsumv += __shfl_xor(wsumv, 16, 32);   // low-half + high-half row groups

    __shared__ float wsum[THREADS / 32];
    const int lane = t & 31, wave = t >> 5;
    if (lane == 0) wsum[wave] = wsumv;
    __syncthreads();
    if (t == 0) {
        float tot = 0.0f;
#pragma unroll
        for (int w = 0; w < THREADS / 32; ++w) tot += wsum[w];
        partials[s * NCHUNK + c] = tot;   // deterministic (fixed order)
    }
}

// ---------------------------------------------------------------------------
// Stage 2: reduce 32 partials per slice, then per-batch row-major first-max
// over pairs i<j of (sum_i - sum_j)^2.  argmax(sum diffs) == argmax(mae
// diffs) since mae = sum/N (positive scale); best=0 + strict '>' reproduces
// the reference's "keep (0,0) unless some variance strictly > 0" rule.
// ---------------------------------------------------------------------------
__global__ void argmax_kernel(const float* __restrict__ partials,
                              int* __restrict__ sel)
{
    const int t = threadIdx.x;
    __shared__ float m[NSLICE];
    if (t < NSLICE) {
        float ssum = 0.0f;
#pragma unroll
        for (int k = 0; k < NCHUNK; ++k) ssum += partials[t * NCHUNK + k];
        m[t] = ssum;
    }
    __syncthreads();
    if (t < BZ) {
        const float* mb = &m[t * NF];
        float best = 0.0f;
        int bi = 0, bj = 0;
        for (int i = 0; i < NF - 1; ++i) {
            for (int j = i + 1; j < NF; ++j) {
                const float d = mb[i] - mb[j];
                const float v = d * d;
                if (v > best) { best = v; bi = i; bj = j; }  // first-max kept
            }
        }
        sel[t]      = bi;   // g1 source focal index for batch t
        sel[BZ + t] = bj;   // g2 source focal index for batch t
    }
}

// ---------------------------------------------------------------------------
// Stage 3: out[s_out] = x_focal[sel[s_out]*bz + (s_out % bz)], float4 copies,
// non-temporal on both sides (pure stream, no reuse).
// ---------------------------------------------------------------------------
__global__ __launch_bounds__(THREADS) void gather_kernel(
    const float* __restrict__ xf, const int* __restrict__ sel,
    float* __restrict__ out)
{
    const size_t o      = (size_t)blockIdx.x * THREADS + threadIdx.x; // f4 idx
    const int    s_out  = (int)(o / CHW4);                            // 0..7
    const size_t within = o - (size_t)s_out * CHW4;
    const int    b      = s_out & (BZ - 1);
    const int    f      = sel[s_out];
    const f4* src = (const f4*)xf + (size_t)(f * BZ + b) * CHW4 + within;
    f4 v = __builtin_nontemporal_load(src);
    __builtin_nontemporal_store(v, (f4*)out + o);
}

// ---------------------------------------------------------------------------
extern "C" void kernel_launch(void* const* d_in, const int* in_sizes, int n_in,
                              void* d_out, int out_size, void* d_ws, size_t ws_size,
                              hipStream_t stream)
{
    const float* x   = (const float*)d_in[0];   // [4, 32, 256, 256]
    const float* xf  = (const float*)d_in[1];   // [48, 32, 256, 256]
    const float* sal = (const float*)d_in[2];   // [4, 1, 256, 256]

    float* partials = (float*)d_ws;                                   // 1536 f32
    int*   sel      = (int*)((char*)d_ws + NSLICE * NCHUNK * sizeof(float)); // 8 ints

    mae_partial_kernel<<<NSLICE * NCHUNK, THREADS, 0, stream>>>(x, xf, sal, partials);
    argmax_kernel<<<1, 64, 0, stream>>>(partials, sel);
    gather_kernel<<<(2 * BZ * CHW4) / THREADS, THREADS, 0, stream>>>(xf, sel, (float*)d_out);
}